// AxialShift_37933151158927
// MI455X (gfx1250) — compile-verified
//
#include <hip/hip_runtime.h>
#include <math.h>

typedef __attribute__((ext_vector_type(2))) float v2f;
typedef __attribute__((ext_vector_type(8))) float v8f;

namespace {
constexpr int C_DIM = 192;
constexpr int RES = 32;
constexpr int NPOS = RES * RES * RES;                       // 32768
constexpr int BATCH = 8;
constexpr int KSTEPS = C_DIM / 4;                           // 48 WMMA K-steps
constexpr int NWAVES = 12;                                  // one 16-row M tile per wave
constexpr int BLOCK_THREADS = NWAVES * 32;                  // 384 (wave32)
constexpr int COLS = 16;                                    // N tile (spatial positions)
constexpr int BLOCKS_PER_SAMPLE = 32;
constexpr int POS_PER_BLOCK = NPOS / BLOCKS_PER_SAMPLE;     // 1024
constexpr int ITERS = POS_PER_BLOCK / COLS;                 // 64
constexpr int LDS_STRIDE = C_DIM + 2;                       // 194: odd-ish bank stagger
constexpr int ELEMS_PER_THREAD = (C_DIM * COLS) / BLOCK_THREADS;  // 8
constexpr size_t TENSOR_ELEMS = (size_t)BATCH * C_DIM * NPOS;     // 50,331,648
}  // namespace

__device__ __forceinline__ float gelu_exact(float v) {
  return 0.5f * v * (1.0f + erff(v * 0.70710678118654752440f));
}

// D(16x16,f32) = A(16x4,f32) x B(4x16,f32) + C  on CDNA5 WMMA.
__device__ __forceinline__ v8f wmma_f32_16x16x4(v2f a, v2f b, v8f c) {
#if __has_builtin(__builtin_amdgcn_wmma_f32_16x16x4_f32)
  return __builtin_amdgcn_wmma_f32_16x16x4_f32(false, a, false, b, (short)0, c,
                                               false, false);
#else
  // Shuffle-based emulation matching the ISA VGPR layouts (fallback only).
  const int lane = (int)(threadIdx.x & 31);
  const int hi = lane >> 4;
  const int n = lane & 15;
#pragma unroll
  for (int v = 0; v < 8; ++v) {
    const int m = v + 8 * hi;
#pragma unroll
    for (int k = 0; k < 4; ++k) {
      float av = __shfl((k & 1) ? a.y : a.x, m + 16 * (k >> 1), 32);
      float bv = __shfl((k & 1) ? b.y : b.x, n + 16 * (k >> 1), 32);
      c[v] += av * bv;
    }
  }
  return c;
#endif
}

// Fused channel-GEMM: out[o,p] = sum_c W[o,c] * f(in[c, shift(p,c)]) + bias[o]
// f = optional groupnorm affine + optional exact-GELU (applied on load path).
// Optional GELU + per-sample sum/sumsq accumulation on the output path.
template <int AXIS, bool GN_IN, bool GELU_IN, bool GELU_OUT, bool STATS>
__global__ __launch_bounds__(BLOCK_THREADS) void fused_gemm192(
    const float* __restrict__ in, const float* __restrict__ W,
    const float* __restrict__ bias, const float* __restrict__ gw,
    const float* __restrict__ gb, const float* __restrict__ murstd,
    float* __restrict__ out, float* __restrict__ stats) {
  __shared__ float Bs[COLS * LDS_STRIDE];  // activations, [n][c] layout
  __shared__ float redS[NWAVES];
  __shared__ float redQ[NWAVES];

  const int tid = (int)threadIdx.x;
  const int lane = tid & 31;
  const int wave = tid >> 5;
  const int hi = lane >> 4;  // 0: K pair {0,1}, 1: K pair {2,3}
  const int lo = lane & 15;
  const int b = (int)blockIdx.x / BLOCKS_PER_SAMPLE;
  const int blk = (int)blockIdx.x % BLOCKS_PER_SAMPLE;
  const size_t boff = (size_t)b * C_DIM * NPOS;

  // Resident A operand: this wave's 16x192 weight slice, packed per WMMA layout.
  const int orow = wave * 16 + lo;
  v2f aw[KSTEPS];
#pragma unroll
  for (int j = 0; j < KSTEPS; ++j) {
    const float* wp = W + orow * C_DIM + 4 * j + 2 * hi;
    aw[j].x = wp[0];
    aw[j].y = wp[1];
  }

  float mu = 0.0f, rstd = 1.0f;
  if constexpr (GN_IN) {
    mu = murstd[b];
    rstd = murstd[BATCH + b];
  }

  const int col_n = tid & 15;   // spatial column this thread stages
  const int crow0 = tid >> 4;   // channel rows c = crow0 + 24*rr
  float regs[ELEMS_PER_THREAD];

  auto load_tile = [&](int base_p) {
#pragma unroll
    for (int rr = 0; rr < ELEMS_PER_THREAD; ++rr) {
      const int c = crow0 + rr * (BLOCK_THREADS >> 4);
      int p = base_p + col_n;
      if constexpr (AXIS >= 2) {
        const int stride = (AXIS == 2) ? RES * RES : (AXIS == 3) ? RES : 1;
        const int q = (AXIS == 2)   ? (p >> 10)
                      : (AXIS == 3) ? ((p >> 5) & (RES - 1))
                                    : (p & (RES - 1));
        const int s = (c >> 6) - 1;  // channel chunk -> shift -1, 0, +1
        int q2 = q;
        if (s < 0) q2 = (q == RES - 1) ? (RES - 2) : (q + 1);
        if (s > 0) q2 = (q == 0) ? 1 : (q - 1);
        p += (q2 - q) * stride;
      }
      regs[rr] = in[boff + (size_t)c * NPOS + (size_t)p];
    }
  };

  float ssum = 0.0f, ssq = 0.0f;
  const int base0 = blk * POS_PER_BLOCK;
  load_tile(base0);

  for (int it = 0; it < ITERS; ++it) {
    const int base_p = base0 + it * COLS;
    __syncthreads();
#pragma unroll
    for (int rr = 0; rr < ELEMS_PER_THREAD; ++rr) {
      const int c = crow0 + rr * (BLOCK_THREADS >> 4);
      float v = regs[rr];
      if constexpr (GN_IN) v = (v - mu) * rstd * gw[c] + gb[c];
      if constexpr (GELU_IN) v = gelu_exact(v);
      Bs[col_n * LDS_STRIDE + c] = v;
    }
    __syncthreads();
    if (it + 1 < ITERS) load_tile(base_p + COLS);  // prefetch next tile

    v8f acc0 = {0.f, 0.f, 0.f, 0.f, 0.f, 0.f, 0.f, 0.f};
    v8f acc1 = {0.f, 0.f, 0.f, 0.f, 0.f, 0.f, 0.f, 0.f};
    const float* brow = &Bs[lo * LDS_STRIDE + 2 * hi];
#pragma unroll
    for (int j = 0; j < KSTEPS; j += 2) {
      v2f b0 = *(const v2f*)(brow + 4 * j);
      v2f b1 = *(const v2f*)(brow + 4 * (j + 1));
      acc0 = wmma_f32_16x16x4(aw[j], b0, acc0);
      acc1 = wmma_f32_16x16x4(aw[j + 1], b1, acc1);
    }

    const int p = base_p + lo;  // D column for this lane
#pragma unroll
    for (int v = 0; v < 8; ++v) {
      const int oo = wave * 16 + v + 8 * hi;  // D row per ISA C/D layout
      float val = acc0[v] + acc1[v] + bias[oo];
      if constexpr (GELU_OUT) val = gelu_exact(val);
      out[boff + (size_t)oo * NPOS + (size_t)p] = val;
      if constexpr (STATS) {
        ssum += val;
        ssq += val * val;
      }
    }
  }

  if constexpr (STATS) {
#pragma unroll
    for (int off = 16; off > 0; off >>= 1) {
      ssum += __shfl_down(ssum, off, 32);
      ssq += __shfl_down(ssq, off, 32);
    }
    if (lane == 0) {
      redS[wave] = ssum;
      redQ[wave] = ssq;
    }
    __syncthreads();
    if (tid == 0) {
      float a = 0.f, q = 0.f;
      for (int i = 0; i < NWAVES; ++i) {
        a += redS[i];
        q += redQ[i];
      }
      atomicAdd(&stats[b], a);
      atomicAdd(&stats[BATCH + b], q);
    }
  }
}

__global__ void zero_stats_kernel(float* ws) {
  if (threadIdx.x < 32) ws[threadIdx.x] = 0.0f;
}

__global__ void finalize_stats_kernel(const float* __restrict__ stats,
                                      float* __restrict__ murstd) {
  const int b = (int)threadIdx.x;
  if (b < BATCH) {
    const float inv_n = 1.0f / (float)((long long)C_DIM * NPOS);
    const float mu = stats[b] * inv_n;
    const float var = stats[BATCH + b] * inv_n - mu * mu;
    murstd[b] = mu;
    murstd[BATCH + b] = rsqrtf(var + 1e-5f);
  }
}

extern "C" void kernel_launch(void* const* d_in, const int* in_sizes, int n_in,
                              void* d_out, int out_size, void* d_ws,
                              size_t ws_size, hipStream_t stream) {
  (void)in_sizes;
  (void)n_in;
  (void)out_size;
  (void)ws_size;

  const float* x = (const float*)d_in[0];
  const float* w1 = (const float*)d_in[1];
  const float* b1 = (const float*)d_in[2];
  const float* n1w = (const float*)d_in[3];
  const float* n1b = (const float*)d_in[4];
  const float* w21 = (const float*)d_in[5];
  const float* b21 = (const float*)d_in[6];
  const float* w22 = (const float*)d_in[7];
  const float* b22 = (const float*)d_in[8];
  const float* w23 = (const float*)d_in[9];
  const float* b23 = (const float*)d_in[10];
  const float* n2w = (const float*)d_in[11];
  const float* n2b = (const float*)d_in[12];
  const float* w3 = (const float*)d_in[13];
  const float* b3 = (const float*)d_in[14];

  float* ws = (float*)d_ws;
  float* st1 = ws;        // [8] sum, [8] sumsq
  float* st2 = ws + 16;   // [8] sum, [8] sumsq
  float* mr1 = ws + 32;   // [8] mu, [8] rstd
  float* mr2 = ws + 48;
  float* bufA = ws + 64;            // 201 MB ping buffer
  float* outp = (float*)d_out;      // doubles as the pong buffer

  const dim3 grid(BATCH * BLOCKS_PER_SAMPLE);
  const dim3 blk(BLOCK_THREADS);

  zero_stats_kernel<<<1, 32, 0, stream>>>(ws);

  // t1 = W1 @ x + b1 ; accumulate gn1 stats
  fused_gemm192<-1, false, false, false, true><<<grid, blk, 0, stream>>>(
      x, w1, b1, nullptr, nullptr, nullptr, bufA, st1);
  finalize_stats_kernel<<<1, 32, 0, stream>>>(st1, mr1);

  // t2 = W22 @ shift_y(gelu(gn1(t1))) + b22
  fused_gemm192<3, true, true, false, false><<<grid, blk, 0, stream>>>(
      bufA, w22, b22, n1w, n1b, mr1, outp, nullptr);

  // t3 = W21 @ shift_z(t2) + b21
  fused_gemm192<2, false, false, false, false><<<grid, blk, 0, stream>>>(
      outp, w21, b21, nullptr, nullptr, nullptr, bufA, nullptr);

  // t4 = gelu(W23 @ shift_x(t3) + b23) ; accumulate gn2 stats
  fused_gemm192<4, false, false, true, true><<<grid, blk, 0, stream>>>(
      bufA, w23, b23, nullptr, nullptr, nullptr, outp, st2);
  finalize_stats_kernel<<<1, 32, 0, stream>>>(st2, mr2);

  // out = W3 @ gn2(t4) + b3   (safe in-place: each position fully read into
  // LDS before any of its outputs are written; blocks own disjoint positions)
  fused_gemm192<-1, true, false, false, false><<<grid, blk, 0, stream>>>(
      outp, w3, b3, n2w, n2b, mr2, outp, nullptr);
}